// CondEyeDistance_62027917689079
// MI455X (gfx1250) — compile-verified
//
#include <hip/hip_runtime.h>
#include <hip/hip_bf16.h>
#include <math.h>

// CondEyeDistance on MI455X (gfx1250, wave32).
// One wave32 == one 25x25 matrix, fully branchless per-lane data path:
//   1) stage outp/inp into zero-padded LDS tiles with coalesced loads
//   2) ip = outp @ inp via V_WMMA_F32_16X16X4_F32 (2x2 tiles of 16x16, K steps of 4),
//      fragments gathered from LDS unconditionally
//   3) fro(ip - I) + L1(outp) folded in; ip stored (incl. zero padding) column-major
//   4) one-sided Jacobi on 25 columns (rows 25..31 are zeros -> no predication)
//   5) global accumulators in d_ws; tiny finalize kernel emits the scalar loss.

typedef __attribute__((ext_vector_type(2))) float v2f;
typedef __attribute__((ext_vector_type(8))) float v8f;

#define BATCH   32768
#define NDIM    25
#define NN      (NDIM * NDIM)     // 625
#define KPAD    28                // K padded to multiple of 4
#define ASTRIDE 28                // A tile: 32 rows x 28 cols
#define BSTRIDE 32                // B tile: 28 rows x 32 cols
#define CSTRIDE 32                // ip tile: column-major, 25 cols x 32 rows (rows 25..31 = 0)
#define WBUF    1792              // floats per wave: A(32*28=896) + B(28*32=896); D (32*32=1024) reuses it
#define WAVES   8                 // matrices per block
#define NSWEEPS 8                 // Jacobi sweeps

__device__ __forceinline__ float wave_sum(float v) {
#pragma unroll
  for (int off = 16; off > 0; off >>= 1) v += __shfl_xor(v, off, 32);
  return v;  // every lane holds the total
}

__global__ void init_ws(float* ws) {
  ws[0] = 0.0f;                               // sum of per-matrix fro norms
  ws[1] = 0.0f;                               // sum of (S-1)^2
  ((unsigned int*)ws)[2] = 0u;                // max S (bits; S >= 0)
  ((unsigned int*)ws)[3] = 0x7F800000u;       // min S (bits) = +inf
  ws[4] = 0.0f;                               // sum |outp|
}

__global__ __launch_bounds__(WAVES * 32)
void cond_main(const float* __restrict__ inp,
               const float* __restrict__ outp,
               float* __restrict__ ws) {
  __shared__ __align__(16) float smem[WAVES * WBUF];

  const int lane = threadIdx.x & 31;
  const int wave = threadIdx.x >> 5;
  const int b    = blockIdx.x * WAVES + wave;   // grid covers BATCH exactly

  const float* Amat = outp + (size_t)b * NN;    // ip = outp @ inp
  const float* Bmat = inp  + (size_t)b * NN;

  float* wv = smem + wave * WBUF;               // this wave's scratch
  float* aP = wv;                               // padded A, stride ASTRIDE
  float* bP = wv + 896;                         // padded B, stride BSTRIDE

  const int half = lane >> 4;                   // 0: lanes 0-15, 1: lanes 16-31
  const int l16  = lane & 15;

  // ---- zero-fill padded staging tiles (16B vector stores, branchless) ----
  {
    float4* z4 = (float4*)wv;
    const float4 zz = make_float4(0.f, 0.f, 0.f, 0.f);
#pragma unroll
    for (int t = 0; t < WBUF / 4 / 32; ++t)     // 448 float4 per wave, 14 per lane
      z4[lane + 32 * t] = zz;
  }

  // ---- coalesced, branchless staging of A (outp) and B (inp) ----
  float l1 = 0.0f;
#pragma unroll
  for (int t = 0; t < 20; ++t) {                // 20*32 = 640 >= 625
    const int idx = lane + 32 * t;
    const int ci  = idx < NN - 1 ? idx : NN - 1;  // clamp; dup lanes rewrite slot (24,24)
    const float va = Amat[ci];
    l1 += (idx < NN) ? fabsf(va) : 0.0f;        // select, not branch
    const int m = ci / NDIM;
    const int k = ci - NDIM * m;
    aP[m * ASTRIDE + k] = va;
  }
#pragma unroll
  for (int t = 0; t < 20; ++t) {
    const int idx = lane + 32 * t;
    const int ci  = idx < NN - 1 ? idx : NN - 1;
    const float vb = Bmat[ci];
    const int k = ci / NDIM;
    const int n = ci - NDIM * k;
    bP[k * BSTRIDE + n] = vb;
  }

  // ---- WMMA: D(32x32) = Apad(32x28) x Bpad(28x32), fp32, K steps of 4 ----
  // A f32 16x4 layout: lanes 0-15 hold K=k0,k0+1 ; lanes 16-31 hold K=k0+2,k0+3 (row = lane&15)
  // B f32 4x16 layout: same K split, col = lane&15.
  v8f acc[2][2] = {};
#pragma unroll
  for (int k0 = 0; k0 < KPAD; k0 += 4) {
    const int kk = k0 + half * 2;
    v2f afrag[2], bfrag[2];
#pragma unroll
    for (int mi = 0; mi < 2; ++mi) {
      const int base = (mi * 16 + l16) * ASTRIDE + kk;   // 8B aligned -> ds_load_b64
      afrag[mi].x = aP[base];
      afrag[mi].y = aP[base + 1];
    }
#pragma unroll
    for (int ni = 0; ni < 2; ++ni) {
      const int col = ni * 16 + l16;
      bfrag[ni].x = bP[kk * BSTRIDE + col];
      bfrag[ni].y = bP[(kk + 1) * BSTRIDE + col];
    }
#pragma unroll
    for (int mi = 0; mi < 2; ++mi)
#pragma unroll
      for (int ni = 0; ni < 2; ++ni)
        acc[mi][ni] = __builtin_amdgcn_wmma_f32_16x16x4_f32(
            /*neg_a=*/false, afrag[mi],
            /*neg_b=*/false, bfrag[ni],
            /*c_mod=*/(short)0, acc[mi][ni],
            /*reuse_a=*/false, /*reuse_b=*/false);
  }

  // ---- fro(ip - I); store ip column-major (pad region stores exact zeros) ----
  // C/D layout: VGPR r -> M = r (lanes 0-15) or r+8 (lanes 16-31); N = lane&15.
  float fsum = 0.0f;
#pragma unroll
  for (int mi = 0; mi < 2; ++mi) {
#pragma unroll
    for (int ni = 0; ni < 2; ++ni) {
#pragma unroll
      for (int r = 0; r < 8; ++r) {
        const int m = mi * 16 + (half ? r + 8 : r);
        const int n = ni * 16 + l16;
        const float v = acc[mi][ni][r];
        const float d = v - ((m == n && m < NDIM) ? 1.0f : 0.0f);  // v==0 in pad -> d==0
        fsum += d * d;
        wv[n * CSTRIDE + m] = v;    // unconditional; rows/cols >= 25 hold zeros
      }
    }
  }
  fsum = wave_sum(fsum);
  l1   = wave_sum(l1);
  if (lane == 0) {
    atomicAdd(&ws[0], sqrtf(fsum));
    atomicAdd(&ws[4], l1);
  }

  // ---- one-sided Jacobi on 25 columns; all 32 lanes run unconditionally ----
  // Rows 25..31 are zero and stay zero (rotation of zeros is zero).
  const int r = lane;
  for (int sweep = 0; sweep < NSWEEPS; ++sweep) {
    for (int i = 0; i < NDIM - 1; ++i) {
      for (int j = i + 1; j < NDIM; ++j) {
        const float ai = wv[i * CSTRIDE + r];
        const float aj = wv[j * CSTRIDE + r];
        const float alpha = wave_sum(ai * ai);
        const float beta  = wave_sum(aj * aj);
        const float gamma = wave_sum(ai * aj);
        if (gamma * gamma > 1e-28f * alpha * beta) {   // wave-uniform scalar branch
          const float zeta = (beta - alpha) / (2.0f * gamma);
          const float t = copysignf(1.0f, zeta) /
                          (fabsf(zeta) + sqrtf(1.0f + zeta * zeta));
          const float c = rsqrtf(1.0f + t * t);
          const float s = c * t;
          wv[i * CSTRIDE + r] = c * ai - s * aj;
          wv[j * CSTRIDE + r] = s * ai + c * aj;
        }
      }
    }
  }

  // ---- singular values = column norms ----
  float ssum = 0.0f, smax = 0.0f, smin = __builtin_inff();
#pragma unroll 5
  for (int j = 0; j < NDIM; ++j) {
    const float cj = wv[j * CSTRIDE + r];
    const float s = sqrtf(wave_sum(cj * cj));
    const float d = s - 1.0f;
    ssum += d * d;
    smax = fmaxf(smax, s);
    smin = fminf(smin, s);
  }
  if (lane == 0) {
    atomicAdd(&ws[1], ssum);
    atomicMax((unsigned int*)ws + 2, __float_as_uint(smax));  // S >= 0: bit order == float order
    atomicMin((unsigned int*)ws + 3, __float_as_uint(smin));
  }
}

__global__ void finalize(const float* __restrict__ ws, float* __restrict__ out) {
  const float maxS = __uint_as_float(((const unsigned int*)ws)[2]);
  const float minS = __uint_as_float(((const unsigned int*)ws)[3]);
  float loss = (ws[0] / (float)BATCH) * 1e-5f                  // INV * mean fro
             + (ws[1] / (float)(BATCH * NDIM)) * 1.0f          // DEV * mean (S-1)^2
             + (logf(maxS) - logf(minS)) * 0.01f               // COND
             + 1e-6f * ws[4];                                  // L1
  out[0] = loss;
}

extern "C" void kernel_launch(void* const* d_in, const int* in_sizes, int n_in,
                              void* d_out, int out_size, void* d_ws, size_t ws_size,
                              hipStream_t stream) {
  const float* inp  = (const float*)d_in[0];
  const float* outp = (const float*)d_in[1];
  float* ws  = (float*)d_ws;
  float* out = (float*)d_out;

  init_ws<<<1, 1, 0, stream>>>(ws);
  cond_main<<<BATCH / WAVES, WAVES * 32, 0, stream>>>(inp, outp, ws);
  finalize<<<1, 1, 0, stream>>>(ws, out);
}